// TensorSquare_63634235457617
// MI455X (gfx1250) — compile-verified
//
#include <hip/hip_runtime.h>
#include <hip/hip_bf16.h>
#include <stdint.h>

// TensorSquare sparse bilinear scatter for MI455X (gfx1250, wave32).
//
//   out[z, rows[e]] += vals[e] * f[z, ci[e]] * f[z, cj[e]]   z in [0,256)
//
// Memory-bound (out ~89MB, fits in 192MB L2). Strategy:
//   kernel 1: zero d_out with b128 stores
//   kernel 2: per block, async-copy a 16-row feature tile (~26KB) into LDS
//             (global_load_async_to_lds_b128 + s_wait_asynccnt, CDNA5 async
//             data path), then grid-stride over nonzeros; each thread reads
//             its (val,row,ci,cj) once and issues 16 global_atomic_add_f32
//             (no-return, RMW resolved in the 192MB L2).

#define TS_BATCH   256
#define TS_ZTILE   16       // batch rows staged in LDS per block
#define TS_THREADS 256      // 8 wave32 waves per workgroup

__global__ void ts_zero_kernel(float4* __restrict__ out, long n4) {
    long i = (long)blockIdx.x * blockDim.x + threadIdx.x;
    long stride = (long)gridDim.x * blockDim.x;
    float4 z = make_float4(0.f, 0.f, 0.f, 0.f);
    for (; i < n4; i += stride) out[i] = z;
}

__device__ __forceinline__ void ts_async_ld_b128(unsigned lds_byte_off,
                                                 const void* gptr) {
    unsigned long long ga = (unsigned long long)(uintptr_t)gptr;
    // vdst = per-lane LDS byte address, vaddr = 64-bit global address, saddr off
    asm volatile("global_load_async_to_lds_b128 %0, %1, off"
                 :: "v"(lds_byte_off), "v"(ga) : "memory");
}

__device__ __forceinline__ void ts_async_ld_b32(unsigned lds_byte_off,
                                                const void* gptr) {
    unsigned long long ga = (unsigned long long)(uintptr_t)gptr;
    asm volatile("global_load_async_to_lds_b32 %0, %1, off"
                 :: "v"(lds_byte_off), "v"(ga) : "memory");
}

__global__ __launch_bounds__(TS_THREADS)
void ts_scatter_kernel(const float* __restrict__ f,      // (256, dim_in)
                       const float* __restrict__ vals,   // (nnz)
                       const int*   __restrict__ rows,   // (nnz)
                       const int*   __restrict__ ci,     // (nnz)
                       const int*   __restrict__ cj,     // (nnz)
                       float*       __restrict__ out,    // (256, dim_out)
                       int nnz, int dim_in, int dim_out) {
    extern __shared__ float lds[];                // TS_ZTILE * dim_in floats

    const int z0 = blockIdx.y * TS_ZTILE;
    const float* src = f + (size_t)z0 * dim_in;   // contiguous ZTILE rows

    // ---- stage feature tile into LDS via CDNA5 async global->LDS copies ----
    const int total  = TS_ZTILE * dim_in;         // floats to stage
    const int total4 = total >> 2;                // float4 transfers
    // base LDS byte offset of the dynamic shared array (flat ptr low 32 bits
    // hold the wave-relative LDS offset on gfx1250)
    const unsigned lds_base = (unsigned)(uintptr_t)&lds[0];
    for (int i = threadIdx.x; i < total4; i += TS_THREADS) {
        ts_async_ld_b128(lds_base + (unsigned)(i << 4), src + (i << 2));
    }
    for (int i = (total4 << 2) + threadIdx.x; i < total; i += TS_THREADS) {
        ts_async_ld_b32(lds_base + (unsigned)(i << 2), src + i);
    }
    asm volatile("s_wait_asynccnt 0" ::: "memory");   // wave's transfers done
    __syncthreads();                                   // whole tile visible

    // ---- grid-stride over nonzeros; reuse indices across the z-tile ----
    const int estride = gridDim.x * TS_THREADS;
    for (int e = blockIdx.x * TS_THREADS + threadIdx.x; e < nnz; e += estride) {
        const float v = vals[e];
        const int   r = rows[e];
        const int   a = ci[e];
        const int   b = cj[e];
        float* op = out + (size_t)z0 * dim_out + r;
        #pragma unroll
        for (int z = 0; z < TS_ZTILE; ++z) {
            const float x = lds[z * dim_in + a];
            const float y = lds[z * dim_in + b];
            // no-return hardware fp32 atomic (global_atomic_add_f32),
            // RMW resolved in L2 (out fits in 192MB L2)
            unsafeAtomicAdd(op + (size_t)z * dim_out, v * x * y);
        }
    }
}

extern "C" void kernel_launch(void* const* d_in, const int* in_sizes, int n_in,
                              void* d_out, int out_size, void* d_ws, size_t ws_size,
                              hipStream_t stream) {
    const float* f    = (const float*)d_in[0];   // (256, dim_in)
    const float* vals = (const float*)d_in[1];   // (nnz)
    const int*   rows = (const int*)  d_in[2];
    const int*   ci   = (const int*)  d_in[3];
    const int*   cj   = (const int*)  d_in[4];
    // d_in[5] is dim_out as a 1-elem device array; derive on host instead:
    const int nnz     = in_sizes[1];
    const int dim_in  = in_sizes[0] / TS_BATCH;
    const int dim_out = out_size   / TS_BATCH;

    // 1) zero the output (out_size = 256*dim_out, divisible by 4)
    {
        long n4 = (long)out_size >> 2;
        int blocks = (int)((n4 + TS_THREADS - 1) / TS_THREADS);
        if (blocks > 4096) blocks = 4096;
        if (blocks < 1) blocks = 1;
        ts_zero_kernel<<<blocks, TS_THREADS, 0, stream>>>((float4*)d_out, n4);
    }

    // 2) scatter: grid.x spreads the nnz stream, grid.y tiles the batch.
    //    Aim for ~8 e-iterations per thread; cap to keep staging traffic low.
    {
        int bx = (nnz + TS_THREADS * 8 - 1) / (TS_THREADS * 8);
        if (bx > 128) bx = 128;
        if (bx < 1)  bx = 1;
        dim3 grid(bx, TS_BATCH / TS_ZTILE, 1);
        size_t shmem = (size_t)TS_ZTILE * dim_in * sizeof(float);  // ~26 KB
        ts_scatter_kernel<<<grid, TS_THREADS, shmem, stream>>>(
            f, vals, rows, ci, cj, (float*)d_out, nnz, dim_in, dim_out);
    }
}